// NETWORK_51780125721509
// MI455X (gfx1250) — compile-verified
//
#include <hip/hip_runtime.h>
#include <math.h>

// One step of a batch-1 plastic RNN (NHID=4096). Bandwidth-bound:
// ~335 MB of HBM traffic after L2 reuse of hebb, ~0.1 GFLOP
// -> ~14.4 us floor at 23.3 TB/s. GEMVs run on the CDNA5 WMMA path
// (v_wmma_f32_16x16x4_f32); broadcast vector chunks are staged into LDS
// with the Tensor Data Mover (tensor_load_to_lds + s_wait_tensorcnt);
// single-use matrix streams use non-temporal loads so hebb (67 MB)
// stays resident in the 192 MB L2 between its two uses.

#define NHID   4096
#define NSLICE 16
#define KSLICE (NHID / NSLICE)   // 256 K-rows per slice

typedef __attribute__((ext_vector_type(2))) float        v2f;
typedef __attribute__((ext_vector_type(4))) float        v4f;
typedef __attribute__((ext_vector_type(8))) float        v8f;
typedef __attribute__((ext_vector_type(4))) unsigned int v4u;
typedef __attribute__((ext_vector_type(8))) int          v8i;
typedef __attribute__((ext_vector_type(4))) int          v4i;

// ---------------------------------------------------------------------------
// TDM: DMA `nelem` contiguous f32 from global into LDS (1-D tile descriptor).
// D# group0: count=1 | lds_addr | global_addr | type=2 ("image")
// D# group1: data_size=4B, tensor_dim0=tile_dim0=nelem, tensor_dim1=tile_dim1=1
// Groups 2/3 zero (<=2-D tensor). amdgpu-toolchain (clang-23) 6-arg builtin:
// (v4u g0, v8i g1, v4i g2, v4i g3, v8i, i32 cpol).
// ---------------------------------------------------------------------------
__device__ inline void tdm_load_f32_row_to_lds(const float* gsrc, unsigned lds_off,
                                               unsigned nelem)
{
    const unsigned long long ga = (unsigned long long)(uintptr_t)gsrc;
    v4u g0;
    g0.x = 1u;                                                // count=1
    g0.y = lds_off;                                           // lds_addr[31:0]
    g0.z = (unsigned)(ga & 0xffffffffu);                      // global_addr[31:0]
    g0.w = (unsigned)((ga >> 32) & 0x01ffffffu) | (2u << 30); // addr[56:32]|type=2
    v8i g1;
    g1[0] = (int)(2u << 16);                         // data_size=2 -> 4 bytes
    g1[1] = (int)((nelem & 0xffffu) << 16);          // tensor_dim0[15:0] @ bits 63:48
    g1[2] = (int)((nelem >> 16) | (1u << 16));       // tensor_dim0 hi | tensor_dim1=1
    g1[3] = (int)((nelem & 0xffffu) << 16);          // tile_dim0 @ bits 127:112
    g1[4] = 1;                                       // tile_dim1=1
    g1[5] = (int)nelem;                              // tensor_dim0_stride[31:0]
    g1[6] = 0;
    g1[7] = 0;
    const v4i z4 = {0, 0, 0, 0};
    const v8i z8 = {0, 0, 0, 0, 0, 0, 0, 0};
    __builtin_amdgcn_tensor_load_to_lds(g0, g1, z4, z4, z8, 0);
}

// ---------------------------------------------------------------------------
// ymid_part[slice][j] = sum_{i in slice} yin[i] * (w[i][j] + alpha[i][j]*hebb[i][j])
// grid (32, NSLICE) x 256: blockIdx.x = 128-column tile, wave = 16-column tile,
// blockIdx.y = K slice. A = yin broadcast into all 16 rows, B = W_eff 4x16 tile.
// w/alpha: non-temporal (single use). hebb: RT so it survives in L2 for reuse.
// ---------------------------------------------------------------------------
__global__ __launch_bounds__(256)
void ymid_gemv_wmma(const float* __restrict__ yin,
                    const float* __restrict__ w,
                    const float* __restrict__ alpha,
                    const float* __restrict__ hebb,
                    float* __restrict__ part)      // [NSLICE][NHID]
{
    __shared__ v4f vchunk[KSLICE / 4];              // 1 KB: yin K-slice

    const int lane = threadIdx.x & 31;
    const int wave = threadIdx.x >> 5;
    const int half = lane >> 4;                     // K pair select (A/B layout)
    const int col  = blockIdx.x * 128 + wave * 16 + (lane & 15);
    const int kb   = blockIdx.y * KSLICE;

    if (threadIdx.x < 32) {                         // wave 0 issues the DMA
        tdm_load_f32_row_to_lds(yin + kb, (unsigned)(uintptr_t)&vchunk[0], KSLICE);
        __builtin_amdgcn_s_wait_tensorcnt(0);
    }
    __syncthreads();

    size_t base = (size_t)(kb + 2 * half) * NHID + col;
    const float* pw = w     + base;
    const float* pa = alpha + base;
    const float* ph = hebb  + base;

    v8f acc0 = {};
    v8f acc1 = {};
    #pragma unroll 4
    for (int s = 0; s < KSLICE / 4; ++s) {
        const v4f yq = vchunk[s];                   // ds_load_b128 broadcast
        v2f a;                                      // A: all 16 rows = yin[k]
        a.x = half ? yq.z : yq.x;
        a.y = half ? yq.w : yq.y;
        const float wx = __builtin_nontemporal_load(pw);
        const float wy = __builtin_nontemporal_load(pw + NHID);
        const float ax = __builtin_nontemporal_load(pa);
        const float ay = __builtin_nontemporal_load(pa + NHID);
        const float hx = ph[0];                     // RT: keep hebb in L2
        const float hy = ph[NHID];
        v2f b;                                      // B: W_eff tile, fused on the fly
        b.x = wx + ax * hx;
        b.y = wy + ay * hy;
        if (s & 1)
            acc1 = __builtin_amdgcn_wmma_f32_16x16x4_f32(false, a, false, b,
                                                         (short)0, acc1, false, false);
        else
            acc0 = __builtin_amdgcn_wmma_f32_16x16x4_f32(false, a, false, b,
                                                         (short)0, acc0, false, false);
        pw += 4 * NHID; pa += 4 * NHID; ph += 4 * NHID;
    }
    const v8f acc = acc0 + acc1;
    if (half == 0)                                  // D row 0: lanes 0-15, VGPR0
        part[(size_t)blockIdx.y * NHID + col] = acc[0];
}

// ---------------------------------------------------------------------------
// ymid[j] = sum over slices (deterministic fixed-order reduction, no atomics)
// ---------------------------------------------------------------------------
__global__ __launch_bounds__(256)
void combine_ymid(const float* __restrict__ part, float* __restrict__ ymid)
{
    const int j = blockIdx.x * blockDim.x + threadIdx.x;
    float s = 0.0f;
    #pragma unroll
    for (int p = 0; p < NSLICE; ++p) s += part[(size_t)p * NHID + j];
    ymid[j] = s;
}

// ---------------------------------------------------------------------------
// eta_part[slice][j] = sum_{i in slice} ymid[i] * pred_eta[i][j]
// (sigmoid + bias applied later in hebb_update). pred_eta is single use -> NT.
// ---------------------------------------------------------------------------
__global__ __launch_bounds__(256)
void eta_gemv_wmma(const float* __restrict__ ymid,
                   const float* __restrict__ pred_eta,
                   float* __restrict__ part)       // [NSLICE][NHID]
{
    __shared__ v4f vchunk[KSLICE / 4];              // 1 KB: ymid K-slice

    const int lane = threadIdx.x & 31;
    const int wave = threadIdx.x >> 5;
    const int half = lane >> 4;
    const int col  = blockIdx.x * 128 + wave * 16 + (lane & 15);
    const int kb   = blockIdx.y * KSLICE;

    if (threadIdx.x < 32) {
        tdm_load_f32_row_to_lds(ymid + kb, (unsigned)(uintptr_t)&vchunk[0], KSLICE);
        __builtin_amdgcn_s_wait_tensorcnt(0);
    }
    __syncthreads();

    const float* pm = pred_eta + (size_t)(kb + 2 * half) * NHID + col;

    v8f acc0 = {};
    v8f acc1 = {};
    #pragma unroll 4
    for (int s = 0; s < KSLICE / 4; ++s) {
        const v4f yq = vchunk[s];
        v2f a;
        a.x = half ? yq.z : yq.x;
        a.y = half ? yq.w : yq.y;
        v2f b;
        b.x = __builtin_nontemporal_load(pm);
        b.y = __builtin_nontemporal_load(pm + NHID);
        if (s & 1)
            acc1 = __builtin_amdgcn_wmma_f32_16x16x4_f32(false, a, false, b,
                                                         (short)0, acc1, false, false);
        else
            acc0 = __builtin_amdgcn_wmma_f32_16x16x4_f32(false, a, false, b,
                                                         (short)0, acc0, false, false);
        pm += 4 * NHID;
    }
    const v8f acc = acc0 + acc1;
    if (half == 0)
        part[(size_t)blockIdx.y * NHID + col] = acc[0];
}

// ---------------------------------------------------------------------------
// Fused: i_t = relu(input@w0); pre = ymid + i_t + b0; LayerNorm; relu;
// yout -> d_out; out = yout@w1 + b1. Single 1024-thread block, LDS reductions.
// ---------------------------------------------------------------------------
__global__ __launch_bounds__(1024)
void stats_kernel(const float* __restrict__ ymid,
                  const float* __restrict__ input,
                  const float* __restrict__ w0,
                  const float* __restrict__ b0,
                  const float* __restrict__ ln_gamma,
                  const float* __restrict__ ln_beta,
                  const float* __restrict__ w1,
                  const float* __restrict__ b1,
                  float* __restrict__ yout,
                  float* __restrict__ out)
{
    __shared__ float red[1024];
    __shared__ float red2[1024];
    __shared__ float sstat[2];

    const int t = threadIdx.x;
    const float in0 = input[0], in1 = input[1], in2 = input[2];

    float pre[4];
    float sum = 0.0f, sumsq = 0.0f;
    #pragma unroll
    for (int c = 0; c < 4; ++c) {
        const int j = t + c * 1024;
        float it = fmaf(in0, w0[j], fmaf(in1, w0[NHID + j], in2 * w0[2 * NHID + j]));
        it = fmaxf(it, 0.0f);
        const float p = ymid[j] + it + b0[j];
        pre[c] = p;
        sum += p;
        sumsq += p * p;
    }
    red[t] = sum; red2[t] = sumsq;
    __syncthreads();
    for (int s = 512; s > 0; s >>= 1) {
        if (t < s) { red[t] += red[t + s]; red2[t] += red2[t + s]; }
        __syncthreads();
    }
    if (t == 0) {
        const float mean = red[0] * (1.0f / NHID);
        const float var  = red2[0] * (1.0f / NHID) - mean * mean;  // biased, torch LN
        sstat[0] = mean;
        sstat[1] = rsqrtf(var + 1e-5f);
    }
    __syncthreads();
    const float mean = sstat[0], rstd = sstat[1];

    float dot = 0.0f;
    #pragma unroll
    for (int c = 0; c < 4; ++c) {
        const int j = t + c * 1024;
        float y = (pre[c] - mean) * rstd * ln_gamma[j] + ln_beta[j];
        y = fmaxf(y, 0.0f);
        yout[j] = y;
        dot += y * w1[j];
    }
    red[t] = dot;
    __syncthreads();
    for (int s = 512; s > 0; s >>= 1) {
        if (t < s) red[t] += red[t + s];
        __syncthreads();
    }
    if (t == 0) out[0] = red[0] + b1[0];
}

// ---------------------------------------------------------------------------
// hebb_new[i][j] = (1-eta)*hebb[i][j] + sigmoid(eta_logit[i]+b[i]) * yin[i] * ymid[j]
// hebb re-read should hit L2 (kept RT in the GEMV); it is last-use here -> NT
// load. b128 nontemporal stores for the write-once 67 MB output.
// grid (4, NHID) x 256: one row per blockIdx.y, 1024 columns per block.
// ---------------------------------------------------------------------------
__global__ __launch_bounds__(256)
void hebb_update(const float* __restrict__ hebb,
                 const float* __restrict__ yin,
                 const float* __restrict__ ymid,
                 const float* __restrict__ eta_part,
                 const float* __restrict__ pred_eta_b,
                 const float* __restrict__ eta,
                 float* __restrict__ hebb_out)
{
    const int row = blockIdx.y;
    const int j4  = blockIdx.x * blockDim.x + threadIdx.x;     // float4 column idx

    float lg = pred_eta_b[row];
    #pragma unroll
    for (int p = 0; p < NSLICE; ++p) lg += eta_part[(size_t)p * NHID + row];
    const float eh    = 1.0f / (1.0f + __expf(-lg));           // sigmoid
    const float scale = eh * yin[row];
    const float keep  = 1.0f - eta[0];

    const v4f h = __builtin_nontemporal_load((const v4f*)hebb +
                                             (size_t)row * (NHID / 4) + j4);
    const v4f m = ((const v4f*)ymid)[j4];
    const v4f o = keep * h + scale * m;
    __builtin_nontemporal_store(o, (v4f*)hebb_out + (size_t)row * (NHID / 4) + j4);
}

// ---------------------------------------------------------------------------
extern "C" void kernel_launch(void* const* d_in, const int* in_sizes, int n_in,
                              void* d_out, int out_size, void* d_ws, size_t ws_size,
                              hipStream_t stream)
{
    (void)in_sizes; (void)n_in; (void)out_size; (void)ws_size;

    const float* input      = (const float*)d_in[0];   // (1,3)
    const float* yin        = (const float*)d_in[1];   // (1,NHID)
    const float* hebb       = (const float*)d_in[2];   // (NHID,NHID)
    const float* w0         = (const float*)d_in[3];   // (3,NHID)
    const float* b0         = (const float*)d_in[4];   // (NHID,)
    const float* w          = (const float*)d_in[5];   // (NHID,NHID)
    const float* w1         = (const float*)d_in[6];   // (NHID,1)
    const float* b1         = (const float*)d_in[7];   // (1,)
    const float* alpha      = (const float*)d_in[8];   // (NHID,NHID)
    const float* eta        = (const float*)d_in[9];   // (1,)
    const float* pred_eta   = (const float*)d_in[10];  // (NHID,NHID)
    const float* pred_eta_b = (const float*)d_in[11];  // (NHID,)
    const float* ln_gamma   = (const float*)d_in[12];  // (NHID,)
    const float* ln_beta    = (const float*)d_in[13];  // (NHID,)

    // Outputs concatenated: yout (NHID), hebb_new (NHID*NHID), out (1)
    float* yout     = (float*)d_out;
    float* hebb_out = yout + NHID;
    float* outp     = hebb_out + (size_t)NHID * NHID;

    // Workspace: ymid partials, combined ymid, eta logit partials (~0.53 MB)
    float* ws        = (float*)d_ws;
    float* ymid_part = ws;                                 // NSLICE*NHID
    float* ymid      = ymid_part + (size_t)NSLICE * NHID;  // NHID
    float* eta_part  = ymid + NHID;                        // NSLICE*NHID

    ymid_gemv_wmma<<<dim3(32, NSLICE), 256, 0, stream>>>(yin, w, alpha, hebb, ymid_part);
    combine_ymid<<<NHID / 256, 256, 0, stream>>>(ymid_part, ymid);
    eta_gemv_wmma<<<dim3(32, NSLICE), 256, 0, stream>>>(ymid, pred_eta, eta_part);
    stats_kernel<<<1, 1024, 0, stream>>>(ymid, input, w0, b0, ln_gamma, ln_beta,
                                         w1, b1, yout, outp);
    hebb_update<<<dim3(4, NHID), 256, 0, stream>>>(hebb, yin, ymid, eta_part,
                                                   pred_eta_b, eta, hebb_out);
}